// Memory_64476049048098
// MI455X (gfx1250) — compile-verified
//
#include <hip/hip_runtime.h>
#include <hip/hip_bf16.h>

typedef float v2f __attribute__((ext_vector_type(2)));
typedef float v8f __attribute__((ext_vector_type(8)));

#define FDIM    256
#define MROWS   512
#define NTILE   64          // columns per workgroup
#define BSTRIDE 260         // 256 + 4 pad floats -> conflict-free b64 LDS reads
#define EPSV    1e-7f

// Order-preserving float -> u32 (monotonic for all non-NaN, incl. +-inf)
__device__ __forceinline__ unsigned fkey(float f) {
    unsigned u = __float_as_uint(f);
    return (u & 0x80000000u) ? ~u : (u | 0x80000000u);
}
__device__ __forceinline__ float funkey(unsigned e) {
    unsigned u = (e & 0x80000000u) ? (e ^ 0x80000000u) : ~e;
    return __uint_as_float(u);
}

// ---- kernel 0: descriptor norms + key init -------------------------------
__global__ __launch_bounds__(256) void init_norms_kernel(
    const float* __restrict__ desc, float* __restrict__ xx,
    unsigned long long* __restrict__ keys)
{
    int r = blockIdx.x * 256 + threadIdx.x;      // 0..511
    const float4* p = (const float4*)(desc + (size_t)r * FDIM);
    float s = 0.f;
#pragma unroll
    for (int j = 0; j < FDIM / 4; ++j) {
        float4 v = p[j];
        s += v.x * v.x + v.y * v.y + v.z * v.z + v.w * v.w;
    }
    xx[r] = sqrtf(s);
    keys[r] = 0ull;                               // below any real candidate key
}

// ---- kernel 1: WMMA cosine + masked argmax via u64 atomic max ------------
__global__ __launch_bounds__(256) void cos_argmax_kernel(
    const float* __restrict__ desc, const float* __restrict__ mem,
    const float* __restrict__ rank, const float* __restrict__ xx,
    unsigned long long* __restrict__ keys, int N)
{
    extern __shared__ float smem[];
    float* sB  = smem;                    // [NTILE][BSTRIDE]
    float* sYY = smem + NTILE * BSTRIDE;  // [NTILE] (negative => masked/invalid)

    const int tid  = threadIdx.x;
    const long long col0 = (long long)blockIdx.x * NTILE;

    // Stage B tile: 4 threads per column, 64 contiguous floats each.
    {
        int c   = tid >> 2;
        int sub = tid & 3;
        long long col = col0 + c;
        float* dst = sB + c * BSTRIDE + sub * 64;
        if (col < N) {
            const float* src = mem + (size_t)col * FDIM + sub * 64;
#pragma unroll
            for (int j = 0; j < 16; ++j)
                *(float4*)(dst + j * 4) = *(const float4*)(src + j * 4);
        } else {
            float4 z = {0.f, 0.f, 0.f, 0.f};
#pragma unroll
            for (int j = 0; j < 16; ++j) *(float4*)(dst + j * 4) = z;
        }
    }
    __syncthreads();

    // Column norms + rank mask (threads 0..63), sentinel -1 when excluded.
    if (tid < NTILE) {
        long long col = col0 + tid;
        float v = -1.0f;
        if (col < N && rank[col] > 0.0f) {
            const float* row = sB + tid * BSTRIDE;
            float s = 0.f;
#pragma unroll 4
            for (int f = 0; f < FDIM; ++f) s += row[f] * row[f];
            v = sqrtf(s);
        }
        sYY[tid] = v;
    }
    __syncthreads();

    const int wave = tid >> 5;
    const int lane = tid & 31;
    const int ln   = lane & 15;
    const int hi   = lane >> 4;                  // K-pair select (A/B), M-half (C)
    const float NEG_INF = -__builtin_huge_valf();

#pragma unroll 1
    for (int mi = 0; mi < 4; ++mi) {
        const int mt = wave + mi * 8;            // 0..31, all distinct across waves
        const float* aptr = desc + (size_t)(mt * 16 + ln) * FDIM + hi * 2;
        const float* bbase = sB + ln * BSTRIDE + hi * 2;

        v8f acc[4];
#pragma unroll
        for (int nt = 0; nt < 4; ++nt) acc[nt] = (v8f){0,0,0,0,0,0,0,0};

#pragma unroll 4
        for (int k = 0; k < FDIM; k += 4) {
            v2f a = *(const v2f*)(aptr + k);     // A 16x4 fragment (f32)
#pragma unroll
            for (int nt = 0; nt < 4; ++nt) {
                v2f b = *(const v2f*)(bbase + nt * 16 * BSTRIDE + k);
                acc[nt] = __builtin_amdgcn_wmma_f32_16x16x4_f32(
                    false, a, false, b, (short)0, acc[nt], false, false);
            }
        }

        // Epilogue: cos + per-row running (max, argmin-index) over 64 cols.
        float xxv[8];
#pragma unroll
        for (int i = 0; i < 8; ++i) xxv[i] = xx[mt * 16 + hi * 8 + i];

        float bestv[8];
        int   besti[8];
#pragma unroll
        for (int i = 0; i < 8; ++i) { bestv[i] = NEG_INF; besti[i] = 0x7FFFFFFF; }

#pragma unroll
        for (int nt = 0; nt < 4; ++nt) {
            float yv = sYY[nt * 16 + ln];
            int colIdx = (int)col0 + nt * 16 + ln;
#pragma unroll
            for (int i = 0; i < 8; ++i) {
                float c = (yv < 0.f) ? NEG_INF
                                     : acc[nt][i] / fmaxf(xxv[i] * yv, EPSV);
                if (c > bestv[i] || (c == bestv[i] && colIdx < besti[i])) {
                    bestv[i] = c; besti[i] = colIdx;
                }
            }
        }

        // Cross-lane reduce over the 16 lanes sharing each row (stays in half).
#pragma unroll
        for (int i = 0; i < 8; ++i) {
            float v = bestv[i]; int ix = besti[i];
#pragma unroll
            for (int off = 8; off >= 1; off >>= 1) {
                float ov = __shfl_xor(v, off, 32);
                int   oi = __shfl_xor(ix, off, 32);
                if (ov > v || (ov == v && oi < ix)) { v = ov; ix = oi; }
            }
            if (ln == 0) {
                int m = mt * 16 + hi * 8 + i;
                unsigned long long key =
                    ((unsigned long long)fkey(v) << 32) |
                    (unsigned long long)(0xFFFFFFFFu - (unsigned)ix);
                atomicMax(keys + m, key);        // global_atomic_max_u64
            }
        }
    }
}

// ---- kernel 2: decode keys, write cos_max/idx, gather winning rows -------
__global__ __launch_bounds__(64) void finalize_kernel(
    const unsigned long long* __restrict__ keys,
    const float* __restrict__ mem, float* __restrict__ out)
{
    int m = blockIdx.x;
    unsigned long long key = keys[m];
    float val = funkey((unsigned)(key >> 32));
    int idx = (int)(0xFFFFFFFFu - (unsigned)(key & 0xFFFFFFFFu));
    if (threadIdx.x == 0) {
        out[m] = val;                       // cos_max
        out[MROWS + m] = (float)idx;        // idx (exact: < 2^24)
    }
    const float4* src = (const float4*)(mem + (size_t)idx * FDIM);
    float4* dst = (float4*)(out + 2 * MROWS + (size_t)m * FDIM);
    dst[threadIdx.x] = src[threadIdx.x];    // 64 thr x 16B = 1 KB row
}

extern "C" void kernel_launch(void* const* d_in, const int* in_sizes, int n_in,
                              void* d_out, int out_size, void* d_ws, size_t ws_size,
                              hipStream_t stream) {
    const float* desc = (const float*)d_in[0];   // [512,256] f32
    const float* mem  = (const float*)d_in[1];   // [N,256]  f32
    const float* rank = (const float*)d_in[2];   // [N]      f32
    float* out = (float*)d_out;
    const int N = in_sizes[2];

    unsigned long long* keys = (unsigned long long*)d_ws;              // 512*8 B
    float* xx = (float*)((char*)d_ws + 4096);                          // 512*4 B

    init_norms_kernel<<<MROWS / 256, 256, 0, stream>>>(desc, xx, keys);

    int nchunks = (N + NTILE - 1) / NTILE;
    size_t lds = (size_t)(NTILE * BSTRIDE + NTILE) * sizeof(float);
    cos_argmax_kernel<<<nchunks, 256, lds, stream>>>(desc, mem, rank, xx, keys, N);

    finalize_kernel<<<MROWS, 64, 0, stream>>>(keys, mem, out);
}